// EGATConv_30846455120538
// MI455X (gfx1250) — compile-verified
//
#include <hip/hip_runtime.h>

#define N_NODES   50000
#define N_EDGES   1600000
#define E_FULL    (N_EDGES + N_NODES)   // 1,650,000 (with self loops)
#define IN_CH     128
#define HEADS     4
#define OUT_CH    16
#define HC        (HEADS * OUT_CH)      // 64
#define NEG_SLOPE 0.2f

typedef __attribute__((ext_vector_type(2))) float v2f;
typedef __attribute__((ext_vector_type(8))) float v8f;

// ---------------------------------------------------------------------------
// Kernel 1: h = x @ weight  (50000x128 @ 128x64, f32) via V_WMMA_F32_16X16X4_F32
// One wave -> one 16(M) x 64(N) output block, K-loop of 32 steps (K=4 each).
// A fragment layout (16x4 f32): lane l, VGPR i -> A[M = l&15][K = 2*(l>>4)+i]
// B fragment layout (4x16 f32): lane l, VGPR i -> B[K = 2*(l>>4)+i][N = l&15]
// Weight staged in LDS transposed (wt[n][k]) so each B fragment is one
// 8-byte ds_load_b64.
// ---------------------------------------------------------------------------
__global__ void __launch_bounds__(256) gemm_xw_wmma(const float* __restrict__ x,
                                                    const float* __restrict__ w,
                                                    float* __restrict__ h) {
  __shared__ float wt[HC * IN_CH];  // 64 x 128 f32 = 32 KB (of 320 KB/WGP)
  for (int i = threadIdx.x; i < IN_CH * HC; i += 256) {
    int k = i >> 6;                 // 0..127
    int n = i & 63;                 // 0..63
    wt[n * IN_CH + k] = w[i];       // w is row-major [k][n]
  }
  __syncthreads();

  const int wave = threadIdx.x >> 5;
  const int lane = threadIdx.x & 31;
  const int m0   = (blockIdx.x * 8 + wave) * 16;
  if (m0 >= N_NODES) return;        // wave-uniform: EXEC stays all-1s for WMMA

  const int mrow  = m0 + (lane & 15);
  const int khalf = (lane >> 4) * 2;    // 0 or 2
  const int ncol  = lane & 15;

  v8f acc0 = {}, acc1 = {}, acc2 = {}, acc3 = {};
  const float* xr = x + (size_t)mrow * IN_CH;

#pragma unroll
  for (int kk = 0; kk < IN_CH; kk += 4) {
    v2f a  = *(const v2f*)(xr + kk + khalf);                          // global_load_b64
    v2f b0 = *(const v2f*)(&wt[( 0 + ncol) * IN_CH + kk + khalf]);    // ds_load_b64
    v2f b1 = *(const v2f*)(&wt[(16 + ncol) * IN_CH + kk + khalf]);
    v2f b2 = *(const v2f*)(&wt[(32 + ncol) * IN_CH + kk + khalf]);
    v2f b3 = *(const v2f*)(&wt[(48 + ncol) * IN_CH + kk + khalf]);
    acc0 = __builtin_amdgcn_wmma_f32_16x16x4_f32(false, a, false, b0, (short)0, acc0, false, false);
    acc1 = __builtin_amdgcn_wmma_f32_16x16x4_f32(false, a, false, b1, (short)0, acc1, false, false);
    acc2 = __builtin_amdgcn_wmma_f32_16x16x4_f32(false, a, false, b2, (short)0, acc2, false, false);
    acc3 = __builtin_amdgcn_wmma_f32_16x16x4_f32(false, a, false, b3, (short)0, acc3, false, false);
  }

  // C/D layout: VGPR j, lane l -> D[M = j + 8*(l>>4)][N = l&15]
  const int mstore = m0 + 8 * (lane >> 4);
#pragma unroll
  for (int j = 0; j < 8; ++j) {
    float* hr = h + (size_t)(mstore + j) * HC + ncol;
    hr[0]  = acc0[j];
    hr[16] = acc1[j];
    hr[32] = acc2[j];
    hr[48] = acc3[j];
  }
}

// ---------------------------------------------------------------------------
// Kernel 2: out = bias (broadcast).  Must rerun every launch (graph replay
// does not re-zero d_out) so the atomic segment-sum starts clean.
// ---------------------------------------------------------------------------
__global__ void init_out(float* __restrict__ out, const float* __restrict__ bias) {
  int i = blockIdx.x * 256 + threadIdx.x;
  if (i < N_NODES * HC) out[i] = bias[i & (HC - 1)];
}

// ---------------------------------------------------------------------------
// Kernel 3: per-edge attention + scatter-add.  One wave per edge.
// Lane covers channel pair (2*lane, 2*lane+1) -> 8 lanes per head.
//   gather h[row], h[col]      : one v2f per lane = one 256B coalesced txn/node
//   per-head dot products      : __shfl_xor tree over 8-lane groups
//   alpha = leakyrelu(.) * ea  ; msg = alpha * h[col] -> atomicAdd(out[row])
// Also emits edge_index_full (int32 bit-exact) and alpha outputs.
// ---------------------------------------------------------------------------
__global__ void __launch_bounds__(256) egat_edge(const float* __restrict__ h,
                                                 const int*   __restrict__ ei,
                                                 const float* __restrict__ eattr,
                                                 const float* __restrict__ att,
                                                 float* __restrict__ out,
                                                 float* __restrict__ alpha_out,
                                                 int*   __restrict__ ei_out) {
  const int e = blockIdx.x * 8 + (threadIdx.x >> 5);
  if (e >= E_FULL) return;          // wave-uniform exit
  const int lane = threadIdx.x & 31;

  int r, c;
  if (e < N_EDGES) { r = ei[e]; c = ei[N_EDGES + e]; }
  else             { r = e - N_EDGES; c = r; }        // self loop

  const int hd = lane >> 3;         // head 0..3
  const int ch = lane * 2;          // first channel of this lane's pair
  const int ci = ch & 15;           // within-head channel index (even)

  v2f hr = *(const v2f*)(h + (size_t)r * HC + ch);
  v2f hc = *(const v2f*)(h + (size_t)c * HC + ch);

  // att_weight[0, hd, 0:16] = a_src ; att_weight[0, hd, 16:32] = a_dst
  const float as0 = att[hd * 32 + ci];
  const float as1 = att[hd * 32 + ci + 1];
  const float ad0 = att[hd * 32 + 16 + ci];
  const float ad1 = att[hd * 32 + 16 + ci + 1];

  float p = hr.x * as0 + hr.y * as1 + hc.x * ad0 + hc.y * ad1;
  // reduce over the 8 lanes of this head (xor masks stay inside the group)
  p += __shfl_xor(p, 1, 32);
  p += __shfl_xor(p, 2, 32);
  p += __shfl_xor(p, 4, 32);

  float alpha = (p >= 0.0f) ? p : NEG_SLOPE * p;
  const float ea = (e < N_EDGES) ? eattr[(size_t)e * HEADS + hd] : 1.0f;
  alpha *= ea;

  // segment_sum(msg, row):  out[r] += alpha * h[col]
  float* dst = out + (size_t)r * HC + ch;
  atomicAdd(dst + 0, alpha * hc.x);
  atomicAdd(dst + 1, alpha * hc.y);

  if ((lane & 7) == 0) alpha_out[(size_t)e * HEADS + hd] = alpha;
  if (lane == 0) { ei_out[e] = r; ei_out[E_FULL + e] = c; }
}

// ---------------------------------------------------------------------------
extern "C" void kernel_launch(void* const* d_in, const int* in_sizes, int n_in,
                              void* d_out, int out_size, void* d_ws, size_t ws_size,
                              hipStream_t stream) {
  const float* x     = (const float*)d_in[0];   // (50000, 128)
  const int*   ei    = (const int*)  d_in[1];   // (2, 1600000)
  const float* eattr = (const float*)d_in[2];   // (1600000, 4)
  const float* w     = (const float*)d_in[3];   // (128, 64)
  const float* att   = (const float*)d_in[4];   // (1, 4, 32)
  const float* bias  = (const float*)d_in[5];   // (64,)

  // d_out = [ out (50000*64 f32) | edge_index_full (2*1650000 i32) | alpha (1650000*4 f32) ]
  float* out_f   = (float*)d_out;
  int*   eif_out = (int*)(out_f + (size_t)N_NODES * HC);
  float* alpha_o = out_f + (size_t)N_NODES * HC + (size_t)2 * E_FULL;

  float* h = (float*)d_ws;                      // 50000*64 f32 = 12.8 MB scratch

  // 1) h = x @ W  (WMMA f32 16x16x4): 3125 row tiles, 8 waves/block
  gemm_xw_wmma<<<(N_NODES / 16 + 7) / 8, 256, 0, stream>>>(x, w, h);

  // 2) out = bias
  init_out<<<(N_NODES * HC + 255) / 256, 256, 0, stream>>>(out_f, bias);

  // 3) edge attention + scatter-add (+ edge_index_full, alpha outputs)
  egat_edge<<<(E_FULL + 7) / 8, 256, 0, stream>>>(h, ei, eattr, att,
                                                  out_f, alpha_o, eif_out);
}